// THB_nn_module_63230508531898
// MI455X (gfx1250) — compile-verified
//
#include <hip/hip_runtime.h>
#include <hip/hip_bf16.h>

typedef unsigned int u32;
typedef __attribute__((ext_vector_type(4))) u32 v4u;
typedef __attribute__((ext_vector_type(8))) int  v8i;
typedef __attribute__((ext_vector_type(4))) int  v4i;

#define TILE   64          // points per block
#define SUPP   64          // supports per point (4^3 trivariate cubic)
#define STRIDE 65          // LDS dword stride per point (64 + 1 TDM pad dword)

// ---------------------------------------------------------------------------
// Fast path: uniform S=64 segments. TDM streams PHI/Jm tiles into LDS with
// per-64-dword padding (stride-65 layout -> conflict-free lane-per-point
// reads). Control-point gathers go straight through L2 (6 MB table, resident).
// ---------------------------------------------------------------------------
__global__ __launch_bounds__(TILE)
void thb_eval_tdm_kernel(const float* __restrict__ ctrl,
                         const float* __restrict__ PHI,
                         const int*   __restrict__ Jm,
                         float*       __restrict__ out)
{
    __shared__ u32 lds_buf[2 * TILE * STRIDE];   // 33,280 bytes

    const u32 tid = threadIdx.x;
    const u32 blk = blockIdx.x;
    const unsigned long long elem_base =
        (unsigned long long)blk * (unsigned long long)(TILE * SUPP);

    if (tid < 32) {  // wave 0 of the block issues the two TDM DMAs
        const u32 lds_phi = (u32)(unsigned long long)(uintptr_t)&lds_buf[0];
        const u32 lds_jm  = lds_phi + (u32)(TILE * STRIDE * 4);
        const unsigned long long ga_phi =
            (unsigned long long)(uintptr_t)(PHI + elem_base);
        const unsigned long long ga_jm =
            (unsigned long long)(uintptr_t)(Jm + elem_base);

        // D# group 0: count=1 (valid), lds_addr, 57-bit global_addr, type=2
        v4u g0_phi = { 1u, lds_phi, (u32)ga_phi,
                       (u32)((ga_phi >> 32) & 0x1FFFFFFu) | (2u << 30) };
        v4u g0_jm  = { 1u, lds_jm,  (u32)ga_jm,
                       (u32)((ga_jm  >> 32) & 0x1FFFFFFu) | (2u << 30) };

        // D# group 1:
        //  [17:16] data_size=2 (4B), [20] pad_enable=1,
        //  [24:22] pad_interval=5 (every 64 dwords), [31:25] pad_amount=0 (1 dword)
        //  tensor_dim0=4096, tensor_dim1=1, tile_dim0=4096 (1-D tile),
        //  tile_dim1/2=0 (unused), tensor_dim0_stride=4096
        v8i g1 = {
            (int)0x01520000,        // mask=0 | data_size | pad fields
            (int)(4096u << 16),     // abar_addr=0 | tensor_dim0[15:0]
            (int)(1u << 16),        // tensor_dim0[31:16]=0 | tensor_dim1[15:0]=1
            (int)(4096u << 16),     // tensor_dim1[31:16]=0 | tile_dim0=4096
            0,                      // tile_dim1=0 | tile_dim2=0
            4096,                   // tensor_dim0_stride[31:0]
            0, 0                    // stride0 hi | tensor_dim1_stride
        };
        // D# groups 2/3: higher dims = 1, tile dims unused
        v4i g2 = { 1, 1, 0, 0 };               // tensor_dim2=1, tensor_dim3=1
        v4i g3 = { 0, (int)(1u << 16), 0, 0 }; // tensor_dim4=1, tile_dim4=0
        // 6-arg toolchain form: extra int32x8 operand (zero-filled, probe-safe)
        v8i gx = { 0, 0, 0, 0, 0, 0, 0, 0 };

        __builtin_amdgcn_tensor_load_to_lds(g0_phi, g1, g2, g3, gx, 0);
        __builtin_amdgcn_tensor_load_to_lds(g0_jm,  g1, g2, g3, gx, 0);
        __builtin_amdgcn_s_wait_tensorcnt(0);
    }
    __syncthreads();

    // Lane-per-point accumulation. LDS bank = (tid + s) mod 64 -> conflict-free.
    const u32* __restrict__ ph = &lds_buf[tid * STRIDE];
    const u32* __restrict__ jm = &lds_buf[TILE * STRIDE + tid * STRIDE];

    float ax = 0.0f, ay = 0.0f, az = 0.0f;
#pragma unroll 8
    for (int s = 0; s < SUPP; ++s) {
        const float phi = __uint_as_float(ph[s]);
        const u32   j   = jm[s];
        const float* __restrict__ c = ctrl + 3ull * (unsigned long long)j;
        ax = fmaf(phi, c[0], ax);
        ay = fmaf(phi, c[1], ay);
        az = fmaf(phi, c[2], az);
    }

    const unsigned long long p = (unsigned long long)blk * TILE + tid;
    out[p * 3 + 0] = ax;
    out[p * 3 + 1] = ay;
    out[p * 3 + 2] = az;
}

// ---------------------------------------------------------------------------
// Generic CSR fallback (non-uniform segments): one thread per point.
// ---------------------------------------------------------------------------
__global__ void thb_eval_generic_kernel(const float* __restrict__ ctrl,
                                        const float* __restrict__ PHI,
                                        const int*   __restrict__ Jm,
                                        const int*   __restrict__ ends,
                                        float*       __restrict__ out,
                                        int num_pts)
{
    const int p = blockIdx.x * blockDim.x + threadIdx.x;
    if (p >= num_pts) return;
    const int b = (p == 0) ? 0 : ends[p - 1];
    const int e = ends[p];
    float ax = 0.0f, ay = 0.0f, az = 0.0f;
    for (int i = b; i < e; ++i) {
        const float phi = PHI[i];
        const long long j3 = 3ll * Jm[i];
        ax = fmaf(phi, ctrl[j3 + 0], ax);
        ay = fmaf(phi, ctrl[j3 + 1], ay);
        az = fmaf(phi, ctrl[j3 + 2], az);
    }
    out[3ll * p + 0] = ax;
    out[3ll * p + 1] = ay;
    out[3ll * p + 2] = az;
}

extern "C" void kernel_launch(void* const* d_in, const int* in_sizes, int n_in,
                              void* d_out, int out_size, void* d_ws, size_t ws_size,
                              hipStream_t stream)
{
    const float* ctrl = (const float*)d_in[0];   // [N_CTRL, 3] f32
    const float* PHI  = (const float*)d_in[1];   // [TOTAL]     f32
    const int*   Jm   = (const int*)d_in[2];     // [TOTAL]     i32
    const int*   ends = (const int*)d_in[3];     // [P] inclusive CSR row ends
    float*       out  = (float*)d_out;           // [P, 3]      f32

    const int total   = in_sizes[1];
    const int num_pts = in_sizes[3];
    const int supp    = (num_pts > 0) ? (total / num_pts) : 0;

    if (supp == SUPP && (num_pts % TILE) == 0 && total == num_pts * SUPP) {
        const int blocks = num_pts / TILE;
        thb_eval_tdm_kernel<<<blocks, TILE, 0, stream>>>(ctrl, PHI, Jm, out);
    } else {
        const int threads = 256;
        const int blocks  = (num_pts + threads - 1) / threads;
        thb_eval_generic_kernel<<<blocks, threads, 0, stream>>>(ctrl, PHI, Jm,
                                                                ends, out, num_pts);
    }
}